// PEER_25391846654048
// MI455X (gfx1250) — compile-verified
//
#include <hip/hip_runtime.h>
#include <math.h>

typedef __attribute__((ext_vector_type(2))) float v2f;
typedef __attribute__((ext_vector_type(8))) float v8f;
typedef __attribute__((ext_vector_type(4))) unsigned int v4u;
typedef __attribute__((ext_vector_type(8))) int v8i;
typedef __attribute__((ext_vector_type(4))) int v4i;

#define BN    4096   // B*N tokens
#define DDIM  512
#define HH    8
#define DKEY  256    // dim per sub-key
#define NKEY  256
#define TOPK  16
#define BKC   16     // GEMM K-chunk staged per TDM transfer

// ---------------------------------------------------------------------------
// keysT[(p*8+h)*256 + d][k] = keys[h][k][p][d]   (coalesced writes)
// ---------------------------------------------------------------------------
__global__ void peer_keys_transpose(const float* __restrict__ keys,
                                    float* __restrict__ keysT) {
  int gid = blockIdx.x * blockDim.x + threadIdx.x;   // < 2*8*256*256 = 1048576
  int k = gid & 255;
  int d = (gid >> 8) & 255;
  int h = (gid >> 16) & 7;
  int p = gid >> 19;
  keysT[gid] = keys[(((size_t)(h * NKEY + k) * 2 + p) * DKEY) + d];
}

// ---------------------------------------------------------------------------
// Issue one TDM 2-D tile load (Tensor Data Mover, ISA ch.8):
//   copies tile_d1 rows x tile_d0 elems (fp32) to the LDS address lds_addr, row stride
//   stride0 elems in global memory. D# packed per ISA 8.3/8.4:
//   g0: count=1 | lds_addr | global_addr[56:0] | type=2
//   g1: data_size=4B, tensor_dim0/1 = tile dims, tile_dim0/1, dim0 stride
// This toolchain exposes the 6-arg builtin:
//   (v4u g0, v8i g1, v4i g2, v4i g3, v8i g4, i32 cpol)
// ---------------------------------------------------------------------------
__device__ __forceinline__ void tdm_load_2d(unsigned int lds_addr,
                                            const void* gptr,
                                            unsigned int tile_d0,
                                            unsigned int tile_d1,
                                            unsigned int stride0_elems) {
  unsigned long long ga = (unsigned long long)(uintptr_t)gptr;
  unsigned long long s0 = (unsigned long long)stride0_elems;
  v4u g0;
  g0.x = 1u;                                            // count=1 (valid user D#)
  g0.y = lds_addr;                                      // LDS byte address
  g0.z = (unsigned int)(ga & 0xFFFFFFFFu);              // global_addr[31:0]
  g0.w = (unsigned int)((ga >> 32) & 0x01FFFFFFu) | 0x80000000u;  // [56:32] | type=2
  v8i g1;
  g1[0] = (int)(2u << 16);                              // data_size = 4 bytes
  g1[1] = (int)((tile_d0 & 0xFFFFu) << 16);             // tensor_dim0[15:0]
  g1[2] = (int)(((tile_d0 >> 16) & 0xFFFFu) |           // tensor_dim0[31:16]
                ((tile_d1 & 0xFFFFu) << 16));           // tensor_dim1[15:0]
  g1[3] = (int)(((tile_d1 >> 16) & 0xFFFFu) |           // tensor_dim1[31:16]
                ((tile_d0 & 0xFFFFu) << 16));           // tile_dim0
  g1[4] = (int)(tile_d1 & 0xFFFFu);                     // tile_dim1 | tile_dim2=0
  g1[5] = (int)(unsigned int)(s0 & 0xFFFFFFFFu);        // dim0_stride[31:0]
  g1[6] = (int)(unsigned int)((s0 >> 32) & 0xFFFFu);    // dim0_stride[47:32]
  g1[7] = 0;
  v4i gz4 = {0, 0, 0, 0};                               // 2-D: groups 2/3 unused
  v8i gz8 = {0, 0, 0, 0, 0, 0, 0, 0};
  __builtin_amdgcn_tensor_load_to_lds(g0, g1, gz4, gz4, gz8, 0);
}

// ---------------------------------------------------------------------------
// fp32 WMMA GEMM, TDM-fed and double-buffered:
//   C = A(MxK) * B(KxN), row-major, batched over z.
// Block = 128 threads (4 waves), 64x64 C tile. Per 16-deep K-chunk, wave 0
// issues two tensor_load_to_lds DMAs (A 64x16, B 16x64) into the spare LDS
// buffer, waits s_wait_tensorcnt(2) so chunk c is resident while chunk c+1
// streams, then all 4 waves run V_WMMA_F32_16X16X4_F32 out of LDS.
// ---------------------------------------------------------------------------
__global__ void peer_gemm_wmma(const float* __restrict__ A,
                               const float* __restrict__ B,
                               float* __restrict__ C,
                               int K, int lda, int ldb, int ldc,
                               int zA, int zB, int zC) {
  __shared__ float As[2][64 * BKC];
  __shared__ float Bs[2][BKC * 64];

  A += (size_t)blockIdx.z * zA;
  B += (size_t)blockIdx.z * zB;
  C += (size_t)blockIdx.z * zC;

  const int tid  = threadIdx.x;
  const int wave = tid >> 5;
  const int lane = tid & 31;
  const int row  = lane & 15;   // M (A frag) / N (B,C frag) position
  const int hi   = lane >> 4;   // K pair select; C rows +8

  const int mBlk = blockIdx.x * 64;
  const int n0   = blockIdx.y * 64;
  const int nChunks = K / BKC;

  // prologue: chunk 0 -> buffer 0
  if (tid == 0) {
    tdm_load_2d((unsigned int)(uintptr_t)&As[0][0],
                A + (size_t)mBlk * lda, BKC, 64, (unsigned int)lda);
    tdm_load_2d((unsigned int)(uintptr_t)&Bs[0][0],
                B + n0, 64, BKC, (unsigned int)ldb);
  }

  v8f acc0 = {}, acc1 = {}, acc2 = {}, acc3 = {};

  for (int c = 0; c < nChunks; ++c) {
    const int buf = c & 1;
    if (tid == 0) {
      if (c + 1 < nChunks) {
        const int kk = (c + 1) * BKC;
        tdm_load_2d((unsigned int)(uintptr_t)&As[buf ^ 1][0],
                    A + (size_t)mBlk * lda + kk, BKC, 64, (unsigned int)lda);
        tdm_load_2d((unsigned int)(uintptr_t)&Bs[buf ^ 1][0],
                    B + (size_t)kk * ldb + n0, 64, BKC, (unsigned int)ldb);
        __builtin_amdgcn_s_wait_tensorcnt((short)2);   // chunk c resident
      } else {
        __builtin_amdgcn_s_wait_tensorcnt((short)0);
      }
    }
    __syncthreads();

    const float* __restrict__ Ab = &As[buf][0];
    const float* __restrict__ Bb = &Bs[buf][0];
#pragma unroll
    for (int k4 = 0; k4 < BKC / 4; ++k4) {
      const int kb = k4 * 4 + 2 * hi;
      v2f a;
      a.x = Ab[(wave * 16 + row) * BKC + kb];
      a.y = Ab[(wave * 16 + row) * BKC + kb + 1];
      const float* b0 = Bb + kb * 64 + row;
      const float* b1 = b0 + 64;
      v2f b;
      b.x = b0[0];  b.y = b1[0];
      acc0 = __builtin_amdgcn_wmma_f32_16x16x4_f32(false, a, false, b, (short)0, acc0, false, false);
      b.x = b0[16]; b.y = b1[16];
      acc1 = __builtin_amdgcn_wmma_f32_16x16x4_f32(false, a, false, b, (short)0, acc1, false, false);
      b.x = b0[32]; b.y = b1[32];
      acc2 = __builtin_amdgcn_wmma_f32_16x16x4_f32(false, a, false, b, (short)0, acc2, false, false);
      b.x = b0[48]; b.y = b1[48];
      acc3 = __builtin_amdgcn_wmma_f32_16x16x4_f32(false, a, false, b, (short)0, acc3, false, false);
    }
    __syncthreads();   // protect buffer before next TDM overwrite
  }

  const int m0 = mBlk + wave * 16;
  float* crow = C + n0 + row;
#pragma unroll
  for (int v = 0; v < 8; v++) {
    size_t r = (size_t)(m0 + v + 8 * hi) * ldc;
    crow[r +  0] = acc0[v];
    crow[r + 16] = acc1[v];
    crow[r + 32] = acc2[v];
    crow[r + 48] = acc3[v];
  }
}

// ---------------------------------------------------------------------------
// Wave-wide argmax (value, index) over 32 lanes; result replicated to all.
// ---------------------------------------------------------------------------
__device__ __forceinline__ void wave_argmax(float& bv, int& bi) {
#pragma unroll
  for (int off = 16; off > 0; off >>= 1) {
    float ov = __shfl_xor(bv, off, 32);
    int   oi = __shfl_xor(bi, off, 32);
    if (ov > bv || (ov == bv && oi < bi)) { bv = ov; bi = oi; }
  }
}

// ---------------------------------------------------------------------------
// One wave per (b,n,h):
//   top-16 of sim[p=0], top-16 of sim[p=1], cartesian 16x16 combine top-16,
//   softmax -> (expert id, weight) x16.
// ---------------------------------------------------------------------------
__global__ void peer_topk_combine(const float* __restrict__ sim,
                                  int* __restrict__ idxOut,
                                  float* __restrict__ wOut) {
  __shared__ float s_sx[8][16], s_sy[8][16];
  __shared__ int   s_ix[8][16], s_iy[8][16];

  const int wave = threadIdx.x >> 5;
  const int lane = threadIdx.x & 31;
  const int wid  = blockIdx.x * 8 + wave;   // < BN*HH = 32768
  const int bn   = wid >> 3;
  const int h    = wid & 7;
  const float NEG = -1e30f;

  // ---- per-sub-key top-16 of 256 ----
  for (int p = 0; p < 2; ++p) {
    const float* rowp = sim + (size_t)bn * 4096 + (p * 8 + h) * 256;
    float v[8];
#pragma unroll
    for (int t = 0; t < 8; t++) v[t] = rowp[lane * 8 + t];

#pragma unroll
    for (int s = 0; s < TOPK; s++) {
      float bv = v[0]; int bt = 0;
#pragma unroll
      for (int t = 1; t < 8; t++) if (v[t] > bv) { bv = v[t]; bt = t; }
      int bi = lane * 8 + bt;
      wave_argmax(bv, bi);
      if (lane == 0) {
        if (p == 0) { s_sx[wave][s] = bv; s_ix[wave][s] = bi; }
        else        { s_sy[wave][s] = bv; s_iy[wave][s] = bi; }
      }
      if ((bi >> 3) == lane) {
#pragma unroll
        for (int t = 0; t < 8; t++) if (t == (bi & 7)) v[t] = NEG;
      }
    }
  }
  __syncthreads();

  // ---- cartesian combine: 256 pair sums, top-16 ----
  float v[8];
#pragma unroll
  for (int t = 0; t < 8; t++) {
    int pr = lane * 8 + t;
    v[t] = s_sx[wave][pr >> 4] + s_sy[wave][pr & 15];
  }

  float sc[16]; int eid[16];
#pragma unroll
  for (int s = 0; s < TOPK; s++) {
    float bv = v[0]; int bt = 0;
#pragma unroll
    for (int t = 1; t < 8; t++) if (v[t] > bv) { bv = v[t]; bt = t; }
    int bi = lane * 8 + bt;
    wave_argmax(bv, bi);
    sc[s]  = bv;
    eid[s] = s_ix[wave][bi >> 4] * NKEY + s_iy[wave][bi & 15];
    if ((bi >> 3) == lane) {
#pragma unroll
      for (int t = 0; t < 8; t++) if (t == (bi & 7)) v[t] = NEG;
    }
  }

  // softmax over the 16 selected scores (sc[0] is the max)
  float m = sc[0], sum = 0.f, e[16];
#pragma unroll
  for (int s = 0; s < TOPK; s++) { e[s] = __expf(sc[s] - m); sum += e[s]; }
  float inv = 1.f / sum;

  if (lane == 0) {
#pragma unroll
    for (int s = 0; s < TOPK; s++) {
      idxOut[(size_t)wid * TOPK + s] = eid[s];
      wOut  [(size_t)wid * TOPK + s] = e[s] * inv;
    }
  }
}

// ---------------------------------------------------------------------------
// One 256-thread block (8 waves) per token:
//   out[token] = sum_{h,k} gelu(x . wdown[e]) * softmax_w * wup[e]
// Wave handles one expert at a time: near-cache prefetch of the wup row,
// float4 loads, wave reduce, exact GELU, register accumulation; cross-wave
// reduction via LDS ds_add_f32 atomics. This is the HBM-bound phase
// (~2 GB of expert-row gathers -> ~90+ us at 23.3 TB/s).
// ---------------------------------------------------------------------------
__global__ void peer_expert_apply(const float* __restrict__ x,
                                  const float* __restrict__ wdown,
                                  const float* __restrict__ wup,
                                  const int* __restrict__ idxIn,
                                  const float* __restrict__ wIn,
                                  float* __restrict__ out) {
  __shared__ float xs[DDIM];
  __shared__ float accs[DDIM];
  const int token = blockIdx.x;
  const int tid   = threadIdx.x;       // 256
  const int wave  = tid >> 5;
  const int lane  = tid & 31;

  xs[tid]         = x[(size_t)token * DDIM + tid];
  xs[tid + 256]   = x[(size_t)token * DDIM + tid + 256];
  accs[tid]       = 0.f;
  accs[tid + 256] = 0.f;
  __syncthreads();

  float acc[16];
#pragma unroll
  for (int t = 0; t < 16; t++) acc[t] = 0.f;

  const float4* xs4 = (const float4*)xs;

  for (int pr = wave; pr < HH * TOPK; pr += 8) {
    size_t slot = (size_t)token * (HH * TOPK) + pr;
    int   e  = idxIn[slot];
    float cf = wIn[slot];
    const float4* wd4 = (const float4*)(wdown + (size_t)e * DDIM);
    const float4* wu4 = (const float4*)(wup   + (size_t)e * DDIM);

    // pull the wup row toward near caches while we do the down-projection
    __builtin_prefetch((const void*)(wu4 + lane * 4), 0, 3);

    float dot = 0.f;
#pragma unroll
    for (int t = 0; t < 4; t++) {
      float4 a = wd4[lane * 4 + t];
      float4 b = xs4[lane * 4 + t];
      dot += a.x * b.x + a.y * b.y + a.z * b.z + a.w * b.w;
    }
#pragma unroll
    for (int off = 16; off > 0; off >>= 1) dot += __shfl_xor(dot, off, 32);

    // exact GELU (reference uses approximate=False)
    float g = 0.5f * dot * (1.f + erff(dot * 0.70710678118654752f));
    float c = g * cf;

#pragma unroll
    for (int t = 0; t < 4; t++) {
      float4 u = wu4[lane * 4 + t];
      acc[t * 4 + 0] += c * u.x;
      acc[t * 4 + 1] += c * u.y;
      acc[t * 4 + 2] += c * u.z;
      acc[t * 4 + 3] += c * u.w;
    }
  }

#pragma unroll
  for (int t = 0; t < 16; t++) atomicAdd(&accs[lane * 16 + t], acc[t]);
  __syncthreads();

  out[(size_t)token * DDIM + tid]       = accs[tid];
  out[(size_t)token * DDIM + tid + 256] = accs[tid + 256];
}

// ---------------------------------------------------------------------------
extern "C" void kernel_launch(void* const* d_in, const int* in_sizes, int n_in,
                              void* d_out, int out_size, void* d_ws, size_t ws_size,
                              hipStream_t stream) {
  (void)in_sizes; (void)n_in; (void)out_size; (void)ws_size;

  const float* x     = (const float*)d_in[0];   // [2,2048,512]
  const float* Wq    = (const float*)d_in[1];   // [512, 4096]
  const float* keys  = (const float*)d_in[2];   // [8,256,2,256]
  const float* wdown = (const float*)d_in[3];   // [65536,512]
  const float* wup   = (const float*)d_in[4];   // [65536,512]
  float* out = (float*)d_out;                   // [2,2048,512]

  float* ws     = (float*)d_ws;
  float* keysT  = ws;                       //  1,048,576 f  (4 MB)  [p,h,d,k]
  float* Weff   = keysT + 1048576;          //  2,097,152 f  (8 MB)  [512,4096]
  float* sim    = Weff  + 2097152;          // 16,777,216 f  (64 MB) [4096,4096]
  float* wOut   = sim   + 16777216;         //    524,288 f  (2 MB)
  int*   idxOut = (int*)(wOut + 524288);    //    524,288 i  (2 MB)

  // 1) transpose keys for coalesced B operand
  hipLaunchKernelGGL(peer_keys_transpose, dim3(4096), dim3(256), 0, stream,
                     keys, keysT);

  // 2) fold keys into the projection: Weff[p,h] = Wq_slice @ keysT[p,h]
  //    16 batched GEMMs, M=512 N=256 K=256
  hipLaunchKernelGGL(peer_gemm_wmma, dim3(8, 4, 16), dim3(128), 0, stream,
                     Wq, keysT, Weff,
                     /*K=*/256, /*lda=*/4096, /*ldb=*/256, /*ldc=*/4096,
                     /*zA=*/256, /*zB=*/65536, /*zC=*/256);

  // 3) sim = x @ Weff  (M=4096 N=4096 K=512) — replaces q-projection + key sim
  hipLaunchKernelGGL(peer_gemm_wmma, dim3(64, 64, 1), dim3(128), 0, stream,
                     x, Weff, sim,
                     /*K=*/512, /*lda=*/512, /*ldb=*/4096, /*ldc=*/4096,
                     0, 0, 0);

  // 4) two-stage top-16 + softmax, one wave per (b,n,h)
  hipLaunchKernelGGL(peer_topk_combine, dim3(4096), dim3(256), 0, stream,
                     sim, idxOut, wOut);

  // 5) gather experts, dot + exact GELU + weighted sum (HBM-bound phase)
  hipLaunchKernelGGL(peer_expert_apply, dim3(4096), dim3(256), 0, stream,
                     x, wdown, wup, idxOut, wOut, out);
}